// Combine_Conv2d_19275813224501
// MI455X (gfx1250) — compile-verified
//
#include <hip/hip_runtime.h>
#include <hip/hip_bf16.h>
#include <math.h>

typedef __attribute__((ext_vector_type(2))) float v2f;
typedef __attribute__((ext_vector_type(8))) float v8f;

#define KKH 5
#define KKW 5
#define DD 25          // C*KH*KW
#define IH 112
#define IW 112
#define OH 108
#define OW 108
#define NP (OH*OW)     // 11664 patches per image
#define OC 64
#define NB 32
#define MTILES (NP/16) // 729
#define NTILES (OC/16) // 4
// waves total = NB*MTILES*NTILES = 93312 ; 8 waves/block -> 11664 blocks

#define LOG2E 1.44269504088896340736f

__global__ __launch_bounds__(256) void rbf_conv2d_wmma(
    const float* __restrict__ gimg, const float* __restrict__ rimg,
    const float* __restrict__ gw,   const float* __restrict__ rw,
    const float* __restrict__ stdp, float* __restrict__ out)
{
    const int lane = threadIdx.x & 31;
    const int wid  = (int)((blockIdx.x * blockDim.x + threadIdx.x) >> 5);
    const int ntile = wid & 3;
    int t = wid >> 2;
    const int mtile = t % MTILES;
    const int b     = t / MTILES;

    const int half = lane >> 4;   // 0: K={4j,4j+1}, 1: K={4j+2,4j+3}
    const int mloc = lane & 15;

    // A-fragment role: this lane gathers patch m's elements
    const int m  = mtile * 16 + mloc;
    const int py = m / OW;
    const int px = m % OW;
    const float* gbase = gimg + (size_t)b * (IH*IW) + py * IW + px;
    const float* rbase = rimg + (size_t)b * (IH*IW) + py * IW + px;

    // B-fragment role: this lane gathers channel n's weights
    const int n = ntile * 16 + mloc;
    const float* gwb = gw + n * DD;
    const float* rwb = rw + n * DD;

    v2f ag[7], ar[7], bg[7], br[7];
    float sag = 0.f, sar = 0.f, sbg = 0.f, sbr = 0.f; // partial squared norms

    #pragma unroll
    for (int j = 0; j < 7; ++j) {
        const int k0 = 4*j + 2*half;
        const int k1 = k0 + 1;
        float g0=0.f, g1=0.f, r0=0.f, r1=0.f;
        float wg0=0.f, wg1=0.f, wr0=0.f, wr1=0.f;
        if (k0 < DD) {
            const int o0 = (k0 / KKW) * IW + (k0 % KKW);
            g0 = gbase[o0]; r0 = rbase[o0];
            wg0 = gwb[k0];  wr0 = rwb[k0];
        }
        if (k1 < DD) {
            const int o1 = (k1 / KKW) * IW + (k1 % KKW);
            g1 = gbase[o1]; r1 = rbase[o1];
            wg1 = gwb[k1];  wr1 = rwb[k1];
        }
        sag += g0*g0 + g1*g1;     sar += r0*r0 + r1*r1;
        sbg += wg0*wg0 + wg1*wg1; sbr += wr0*wr0 + wr1*wr1;
        ag[j].x = -2.f*g0; ag[j].y = -2.f*g1;
        ar[j].x = -2.f*r0; ar[j].y = -2.f*r1;
        bg[j].x =  wg0;    bg[j].y =  wg1;
        br[j].x =  wr0;    br[j].y =  wr1;
    }

    // complete squared norms across the two lane halves (wave32)
    sag += __shfl_xor(sag, 16, 32);
    sar += __shfl_xor(sar, 16, 32);
    sbg += __shfl_xor(sbg, 16, 32);
    sbr += __shfl_xor(sbr, 16, 32);

    // augment last k-step, branchless (keeps wave converged before WMMA):
    //   half==0 slots: K=24 (real, keep .x), K=25 -> A=a2, B=1
    //   half==1 slots: K=26 -> A=1, B=w2 ; K=27 -> 0
    const bool h1 = (half != 0);
    ag[6].x = h1 ? 1.f : ag[6].x;   ag[6].y = h1 ? 0.f : sag;
    ar[6].x = h1 ? 1.f : ar[6].x;   ar[6].y = h1 ? 0.f : sar;
    bg[6].x = h1 ? sbg : bg[6].x;   bg[6].y = h1 ? 0.f : 1.f;
    br[6].x = h1 ? sbr : br[6].x;   br[6].y = h1 ? 0.f : 1.f;

    v8f accg = {};
    v8f accr = {};
    #pragma unroll
    for (int j = 0; j < 7; ++j) {
        accg = __builtin_amdgcn_wmma_f32_16x16x4_f32(
            false, ag[j], false, bg[j], (short)0, accg, false, false);
        accr = __builtin_amdgcn_wmma_f32_16x16x4_f32(
            false, ar[j], false, br[j], (short)0, accr, false, false);
    }

    // rbf = exp(-dist^2/(2 s^2)) = exp2(inv * dist^2), inv = -log2(e)/(2 s^2)
    const float s   = stdp[0];
    const float inv = -LOG2E * __builtin_amdgcn_rcpf(2.f * s * s);

    // C/D layout: lanes 0-15 -> M=r (N=lane); lanes 16-31 -> M=r+8 (N=lane-16)
    float res[8];
    #pragma unroll
    for (int r = 0; r < 8; ++r) {
        const float dg = __builtin_amdgcn_sqrtf(__builtin_fmaxf(accg[r], 0.f));
        const float dr = __builtin_amdgcn_sqrtf(__builtin_fmaxf(accr[r], 0.f));
        const float dist = dg + dr;
        res[r] = __builtin_amdgcn_exp2f(inv * dist * dist);
    }

    float* obase = out + ((size_t)(b * OC + n)) * NP + mtile * 16 + half * 8;
    *(float4*)(obase + 0) = *(const float4*)(res + 0);
    *(float4*)(obase + 4) = *(const float4*)(res + 4);
}

extern "C" void kernel_launch(void* const* d_in, const int* in_sizes, int n_in,
                              void* d_out, int out_size, void* d_ws, size_t ws_size,
                              hipStream_t stream) {
    (void)in_sizes; (void)n_in; (void)out_size; (void)d_ws; (void)ws_size;
    const float* gimg = (const float*)d_in[0];
    const float* rimg = (const float*)d_in[1];
    const float* gw   = (const float*)d_in[2];
    const float* rw   = (const float*)d_in[3];
    const float* stdp = (const float*)d_in[4];
    float* out = (float*)d_out;

    dim3 grid(NB * MTILES * NTILES / 8);  // 11664 blocks of 8 waves
    dim3 block(256);
    hipLaunchKernelGGL(rbf_conv2d_wmma, grid, block, 0, stream,
                       gimg, rimg, gw, rw, stdp, out);
}